// LambdaRankLoss_36532991819991
// MI455X (gfx1250) — compile-verified
//
#include <hip/hip_runtime.h>

typedef __attribute__((ext_vector_type(2))) float v2f;
typedef __attribute__((ext_vector_type(8))) float v8f;

#define NB 32               // batch rows
#define LL 1024             // list length
#define TPB 256             // threads per block (8 waves on wave32)
#define WAVES_PER_BLOCK 8
#define CHUNKS (LL / 16)                      // 64 i-chunks of 16 rows per list
#define CHUNK_BLOCKS (CHUNKS / WAVES_PER_BLOCK) // 8 blocks in y per list

// One block per (list b, group of 8 i-chunks). The whole list lives in LDS.
// Phase 1: stable descending ranks of predictions -> per-item discount.
// Phase 2: stable descending ranks of gains -> ideal DCG (block reduction).
// Phase 3: each wave sweeps all j for its 16-row i-chunk; pair values are
//          accumulated into exact-f32 row sums via V_WMMA_F32_16X16X4_F32
//          with a ones B matrix (WMMA as a cross-lane row-reducer).
__global__ __launch_bounds__(TPB)
void lambdarank_rowpair_kernel(const float* __restrict__ pred,
                               const float* __restrict__ tgt,
                               float* __restrict__ partial) {
  __shared__ float sp[LL];   // predictions
  __shared__ float sg[LL];   // gains = 2^t - 1
  __shared__ float sd[LL];   // per-item discount 1/log2(rank+2)
  __shared__ float red[TPB];
  __shared__ float sIdeal;

  const int b = blockIdx.x;
  const int t = threadIdx.x;

  for (int i = t; i < LL; i += TPB) {
    sp[i] = pred[b * LL + i];
    sg[i] = exp2f(tgt[b * LL + i]) - 1.0f;
  }
  __syncthreads();

  // Phase 1: rank[i] = #{j : p_j > p_i} + #{j < i : p_j == p_i}  (stable desc)
  for (int i = t; i < LL; i += TPB) {
    const float pi = sp[i];
    int r = 0;
    for (int j = 0; j < LL; ++j) {
      const float pj = sp[j];
      r += ((pj > pi) || ((pj == pi) && (j < i))) ? 1 : 0;
    }
    sd[i] = 1.0f / log2f((float)(r + 2));
  }

  // Phase 2: ideal DCG via stable descending ranks of gains
  float id = 0.0f;
  for (int i = t; i < LL; i += TPB) {
    const float gi = sg[i];
    int r = 0;
    for (int j = 0; j < LL; ++j) {
      const float gj = sg[j];
      r += ((gj > gi) || ((gj == gi) && (j < i))) ? 1 : 0;
    }
    id += gi / log2f((float)(r + 2));
  }
  red[t] = id;
  __syncthreads();
  for (int off = TPB / 2; off > 0; off >>= 1) {
    if (t < off) red[t] += red[t + off];
    __syncthreads();
  }
  if (t == 0) sIdeal = red[0];
  __syncthreads();   // also publishes all sd[] writes

  // Phase 3: pairwise sweep. sigmoid cancels algebraically:
  //   lambda[i] = -(1/ideal) * sum_j [mis(i,j)] * |g_i-g_j| * |d_i-d_j|
  const int wave = t >> 5;
  const int lane = t & 31;
  const int m    = lane & 15;   // A-matrix row M (both lane halves)
  const int half = lane >> 4;   // 0 -> K=0,1 ; 1 -> K=2,3
  const int chunk = blockIdx.y * WAVES_PER_BLOCK + wave;
  const int i0 = chunk * 16;
  const int i  = i0 + m;

  const float gi = sg[i];
  const float di = sd[i];
  const float pi = sp[i];

  v2f bones; bones.x = 1.0f; bones.y = 1.0f;   // B = ones(4x16)
  v8f acc = {};
  for (int j0 = 0; j0 < LL; j0 += 4) {
    const int j = j0 + 2 * half;
    v2f a;
    {
      const float gj = sg[j], pj = sp[j], dj = sd[j];
      const float gd = gi - gj, pd = pi - pj;
      const bool mis = ((gd > 0.0f) && (pd < 0.0f)) || ((gd < 0.0f) && (pd > 0.0f));
      a.x = mis ? fabsf(gd) * fabsf(di - dj) : 0.0f;
    }
    {
      const float gj = sg[j + 1], pj = sp[j + 1], dj = sd[j + 1];
      const float gd = gi - gj, pd = pi - pj;
      const bool mis = ((gd > 0.0f) && (pd < 0.0f)) || ((gd < 0.0f) && (pd > 0.0f));
      a.y = mis ? fabsf(gd) * fabsf(di - dj) : 0.0f;
    }
    // D[m,n] = sum_k A[m,k] + C[m,n]: every column of C holds the row sum.
    acc = __builtin_amdgcn_wmma_f32_16x16x4_f32(false, a, false, bones,
                                                (short)0, acc, false, false);
  }

  // C/D layout: VGPR v holds M=v (lanes 0-15) and M=v+8 (lanes 16-31),
  // replicated across the 16 columns. Lane 0 owns rows 0..7, lane 16 rows 8..15.
  float s = 0.0f;
  if (m == 0) {
    const int base = i0 + half * 8;
#pragma unroll
    for (int v = 0; v < 8; ++v) s += sp[base + v] * acc[v];
  }
  s += __shfl_xor(s, 16, 32);
  if (lane == 0) {
    const float scale = (sIdeal > 0.0f) ? (-1.0f / sIdeal) : 0.0f;
    partial[b * CHUNKS + chunk] = s * scale;
  }
}

// Deterministic final reduction: fixed mapping + fixed tree order.
__global__ __launch_bounds__(TPB)
void lambdarank_reduce_kernel(const float* __restrict__ partial,
                              float* __restrict__ out) {
  __shared__ float red[TPB];
  const int t = threadIdx.x;
  float s = 0.0f;
  for (int idx = t; idx < NB * CHUNKS; idx += TPB) s += partial[idx];
  red[t] = s;
  __syncthreads();
  for (int off = TPB / 2; off > 0; off >>= 1) {
    if (t < off) red[t] += red[t + off];
    __syncthreads();
  }
  if (t == 0) out[0] = red[0] / (float)NB;
}

extern "C" void kernel_launch(void* const* d_in, const int* in_sizes, int n_in,
                              void* d_out, int out_size, void* d_ws, size_t ws_size,
                              hipStream_t stream) {
  (void)in_sizes; (void)n_in; (void)out_size; (void)ws_size;
  const float* pred = (const float*)d_in[0];
  const float* tgt  = (const float*)d_in[1];
  float* out      = (float*)d_out;
  float* partial  = (float*)d_ws;   // NB*CHUNKS = 2048 floats = 8 KB

  dim3 grid(NB, CHUNK_BLOCKS);
  lambdarank_rowpair_kernel<<<grid, TPB, 0, stream>>>(pred, tgt, partial);
  lambdarank_reduce_kernel<<<1, TPB, 0, stream>>>(partial, out);
}